// PVMLayer3_84327387890624
// MI455X (gfx1250) — compile-verified
//
// PVMLayer3 for MI455X (gfx1250), fp32 end-to-end.
// GEMMs use V_WMMA_F32_16X16X4_F32 (wave32 WMMA). All 1x1 convs / projections
// are NHWC GEMMs through one generic batched WMMA kernel. A and B tiles are
// both staged in LDS with K-innermost padded-18 layout so every WMMA operand
// is a single 8B-aligned ds_load_b64 (no register repacking in the hot loop).
#include <hip/hip_runtime.h>
#include <hip/hip_bf16.h>
#include <math.h>

typedef float v2f __attribute__((ext_vector_type(2)));
typedef float v8f __attribute__((ext_vector_type(8)));

#define BB 8            // batch
#define LL 4096         // 64*64 spatial
#define CC 512          // model channels
#define MB 32           // mamba batch (4 chunks * 8)
#define DI 256          // d_inner
#define NS 16           // d_state
#define DTR 8           // dt_rank
#define DBL_LD 48       // padded ld for [dt|B|C] (=40 real)

__device__ __forceinline__ v8f wmma4_f32(v2f a, v2f b, v8f c) {
  // 8-arg pattern: (neg_a, A, neg_b, B, c_mod, C, reuse_a, reuse_b)
  return __builtin_amdgcn_wmma_f32_16x16x4_f32(false, a, false, b, (short)0, c,
                                               false, false);
}

// ---------------------------------------------------------------------------
// Generic batched GEMM:  C = A(M,K) * W(N,K)^T  [+bias][+C][+s*skip]
// block 128 threads / 4 waves, tile 64(M)x64(N), BK=16.
// Batch pointer = base + (z%bmod)*S1 + (z/bmod)*S2 for A, C, skip.
// ---------------------------------------------------------------------------
__global__ __launch_bounds__(128) void gemm_wmma_f32(
    const float* __restrict__ A, long aS1, long aS2, int bmod, int lda,
    const float* __restrict__ W, int ldw,
    float* __restrict__ C, long cS1, long cS2, int ldc,
    const float* __restrict__ bias,
    const float* __restrict__ skip, long sS1, long sS2, int ldskip,
    const float* __restrict__ skipScale,
    int M, int N, int K, int accumulate) {
  int zb = blockIdx.z;
  int b1 = zb % bmod, b2 = zb / bmod;
  const float* Ab = A + (long)b1 * aS1 + (long)b2 * aS2;
  float* Cb = C + (long)b1 * cS1 + (long)b2 * cS2;
  const float* Sb = skip ? (skip + (long)b1 * sS1 + (long)b2 * sS2) : nullptr;

  // Both tiles K-innermost, padded to 18: K-pairs are 8B aligned (b64 ds
  // loads) and the 16-lane read pattern (ml*18 + const) is bank-conflict-free.
  __shared__ float As[64][18];  // As[m][k]
  __shared__ float Bs[64][18];  // Bs[n][k]  (transposed vs classic layout)

  int tid = threadIdx.x;
  int wave = tid >> 5;
  int lane = tid & 31;
  int ml = lane & 15;          // M (for A) / N (for B) index within 16
  int kb = (lane >> 4) << 1;   // 0 or 2 : K sub-offset per ISA layout
  int m0 = blockIdx.y * 64;
  int n0 = blockIdx.x * 64;

  v8f acc[4];
  v8f zero = {0.f, 0.f, 0.f, 0.f, 0.f, 0.f, 0.f, 0.f};
#pragma unroll
  for (int i = 0; i < 4; ++i) acc[i] = zero;

  for (int k0 = 0; k0 < K; k0 += 16) {
    {  // stage A tile 64x16 (M,K both multiples of 16/64 by construction)
      int r = tid >> 1, c0 = (tid & 1) * 8;
      const float* src = Ab + (long)(m0 + r) * lda + (k0 + c0);
#pragma unroll
      for (int j = 0; j < 8; ++j) As[r][c0 + j] = src[j];
    }
    {  // stage B tile: Bs[n][k] = W[(n0+n)*ldw + k0+k]  (coalesced along k)
      int n = tid >> 1, c0 = (tid & 1) * 8;
      int gn = n0 + n;
      if (gn < N) {
        const float* src = W + (long)gn * ldw + (k0 + c0);
#pragma unroll
        for (int j = 0; j < 8; ++j) Bs[n][c0 + j] = src[j];
      } else {
#pragma unroll
        for (int j = 0; j < 8; ++j) Bs[n][c0 + j] = 0.f;
      }
    }
    __syncthreads();
#pragma unroll
    for (int kk = 0; kk < 16; kk += 4) {
      v2f a;
      a.x = As[wave * 16 + ml][kk + kb];
      a.y = As[wave * 16 + ml][kk + kb + 1];
#pragma unroll
      for (int nt = 0; nt < 4; ++nt) {
        v2f b;
        b.x = Bs[nt * 16 + ml][kk + kb];
        b.y = Bs[nt * 16 + ml][kk + kb + 1];
        acc[nt] = wmma4_f32(a, b, acc[nt]);
      }
    }
    __syncthreads();
  }
  // epilogue: C/D layout -> vgpr r holds rows r (lanes 0-15) and r+8 (16-31)
  float ss = (Sb && skipScale) ? skipScale[0] : 1.f;
  int hi = lane >> 4;
#pragma unroll
  for (int nt = 0; nt < 4; ++nt) {
    int n = n0 + nt * 16 + ml;
    if (n >= N) continue;
    float bv = bias ? bias[n] : 0.f;
#pragma unroll
    for (int r = 0; r < 8; ++r) {
      int m = m0 + wave * 16 + r + 8 * hi;
      long idx = (long)m * ldc + n;
      float v = acc[nt][r] + bv;
      if (accumulate) v += Cb[idx];
      if (Sb) v += ss * Sb[(long)m * ldskip + n];
      Cb[idx] = v;
    }
  }
}

// --------------------------- elementwise / misc ----------------------------
__global__ void k_upsample(const float* __restrict__ x, float* __restrict__ cn) {
  long i = (long)blockIdx.x * blockDim.x + threadIdx.x;  // B*L*C
  if (i >= (long)BB * LL * CC) return;
  int ch = (int)(i & (CC - 1));
  long t = i >> 9;
  int l = (int)(t & (LL - 1));
  int b = (int)(t >> 12);
  int h = l >> 6, w = l & 63;
  cn[i] = x[((long)b * CC + ch) * 1024 + (h >> 1) * 32 + (w >> 1)];
}

__global__ __launch_bounds__(256) void k_seavg(const float* __restrict__ x,
                                               float* __restrict__ avg) {
  int bc = blockIdx.x;  // B*C
  const float* p = x + (long)bc * 1024;
  float s = 0.f;
  for (int i = threadIdx.x; i < 1024; i += 256) s += p[i];
  __shared__ float red[256];
  red[threadIdx.x] = s;
  __syncthreads();
  for (int st = 128; st > 0; st >>= 1) {
    if (threadIdx.x < st) red[threadIdx.x] += red[threadIdx.x + st];
    __syncthreads();
  }
  if (threadIdx.x == 0) avg[bc] = red[0] * (1.f / 1024.f);
}

__global__ __launch_bounds__(512) void k_segate(const float* __restrict__ avg,
                                                const float* __restrict__ w1,
                                                const float* __restrict__ w2,
                                                float* __restrict__ gate) {
  __shared__ float a[CC];
  __shared__ float s1[32];
  int b = blockIdx.x;
  a[threadIdx.x] = avg[b * CC + threadIdx.x];
  __syncthreads();
  if (threadIdx.x < 32) {
    float s = 0.f;
    const float* wr = w1 + threadIdx.x * CC;
    for (int k = 0; k < CC; ++k) s += wr[k] * a[k];
    s1[threadIdx.x] = fmaxf(s, 0.f);
  }
  __syncthreads();
  float s = 0.f;
  const float* wr = w2 + threadIdx.x * 32;
  for (int k = 0; k < 32; ++k) s += wr[k] * s1[k];
  gate[b * CC + threadIdx.x] = 1.f / (1.f + expf(-s));
}

// LayerNorm over C=512; optional per-(b,ch) gate multiplier on the input.
__global__ __launch_bounds__(256) void k_layernorm(
    const float* __restrict__ in, const float* __restrict__ gate,
    const float* __restrict__ g, const float* __restrict__ bt,
    float* __restrict__ out) {
  long row = blockIdx.x;  // b*L + l
  int b = (int)(row >> 12);
  int t = threadIdx.x;
  const float* pin = in + row * CC;
  float v0 = pin[t], v1 = pin[t + 256];
  if (gate) {
    v0 *= gate[b * CC + t];
    v1 *= gate[b * CC + t + 256];
  }
  __shared__ float rs[256], rq[256];
  rs[t] = v0 + v1;
  rq[t] = v0 * v0 + v1 * v1;
  __syncthreads();
  for (int st = 128; st > 0; st >>= 1) {
    if (t < st) {
      rs[t] += rs[t + st];
      rq[t] += rq[t + st];
    }
    __syncthreads();
  }
  float mu = rs[0] * (1.f / CC);
  float var = rq[0] * (1.f / CC) - mu * mu;
  float inv = rsqrtf(var + 1e-5f);
  out[row * CC + t] = (v0 - mu) * inv * g[t] + bt[t];
  out[row * CC + t + 256] = (v1 - mu) * inv * g[t + 256] + bt[t + 256];
}

// causal depthwise conv1d (k=4) + silu; reads xpart (cols 0..255 of xz, ld 512)
__global__ void k_mconv(const float* __restrict__ xz, const float* __restrict__ cw,
                        const float* __restrict__ cb, float* __restrict__ xc) {
  long i = (long)blockIdx.x * blockDim.x + threadIdx.x;  // MB*LL*DI
  if (i >= (long)MB * LL * DI) return;
  int d = (int)(i & (DI - 1));
  long t = i >> 8;
  int l = (int)(t & (LL - 1));
  int mb = (int)(t >> 12);
  const float* base = xz + (long)mb * LL * CC + d;  // ld 512
  float s = cb[d];
#pragma unroll
  for (int j = 0; j < 4; ++j) {
    int ll = l - 3 + j;
    if (ll >= 0) s += cw[d * 4 + j] * base[(long)ll * CC];
  }
  xc[i] = s / (1.f + expf(-s));  // silu
}

// Selective scan: 1 thread per (mb, d). Fuses delta=softplus(dt.w+b) and the
// output gating y'=(y + x*D)*silu(z). Writes into the dead xpart columns.
__global__ __launch_bounds__(256) void k_scan(
    const float* __restrict__ xc, const float* __restrict__ dbl,
    const float* __restrict__ xz, const float* __restrict__ dtw,
    const float* __restrict__ dtb, const float* __restrict__ Alog,
    const float* __restrict__ Dp, float* __restrict__ Y) {
  int mb = blockIdx.x;
  int d = threadIdx.x;
  float wdt[DTR];
#pragma unroll
  for (int j = 0; j < DTR; ++j) wdt[j] = dtw[d * DTR + j];
  float Am[NS];
#pragma unroll
  for (int n = 0; n < NS; ++n) Am[n] = -expf(Alog[d * NS + n]);
  float h[NS];
#pragma unroll
  for (int n = 0; n < NS; ++n) h[n] = 0.f;
  float Dd = Dp[d], dtbd = dtb[d];
  const float* dblb = dbl + (long)mb * LL * DBL_LD;
  const float* xcb = xc + (long)mb * LL * DI + d;
  const float* zb = xz + (long)mb * LL * CC + DI + d;  // zpart column
  float* yb = Y + (long)mb * LL * CC + d;              // overwrite xpart column
  __shared__ float sh[40];
  for (int l = 0; l < LL; ++l) {
    if (threadIdx.x < 40) sh[threadIdx.x] = dblb[(long)l * DBL_LD + threadIdx.x];
    __syncthreads();
    float dtv = dtbd;
#pragma unroll
    for (int j = 0; j < DTR; ++j) dtv += wdt[j] * sh[j];
    float delta = (dtv > 20.f) ? dtv : log1pf(expf(dtv));
    float xt = xcb[(long)l * DI];
    float dx = delta * xt;
    float y = 0.f;
#pragma unroll
    for (int n = 0; n < NS; ++n) {
      float dA = expf(delta * Am[n]);
      h[n] = dA * h[n] + dx * sh[DTR + n];
      y += h[n] * sh[DTR + NS + n];
    }
    float zv = zb[(long)l * CC];
    float yf = (y + xt * Dd) * (zv / (1.f + expf(-zv)));
    yb[(long)l * CC] = yf;
    __syncthreads();
  }
}

__global__ void k_fnhwc(const float* __restrict__ f, float* __restrict__ fn) {
  long i = (long)blockIdx.x * blockDim.x + threadIdx.x;  // B*L*320
  if (i >= (long)BB * LL * 320) return;
  int ch = (int)(i % 320);
  long t = i / 320;
  int l = (int)(t & (LL - 1));
  int b = (int)(t >> 12);
  fn[i] = f[((long)b * 320 + ch) * LL + l];
}

__global__ void k_dwconv(const float* __restrict__ in, const float* __restrict__ w,
                         const float* __restrict__ bias, float* __restrict__ out) {
  long i = (long)blockIdx.x * blockDim.x + threadIdx.x;  // B*L*80
  if (i >= (long)BB * LL * 80) return;
  int ch = (int)(i % 80);
  long t = i / 80;
  int l = (int)(t & (LL - 1));
  int b = (int)(t >> 12);
  int h = l >> 6, wq = l & 63;
  float s = bias[ch];
#pragma unroll
  for (int ky = 0; ky < 3; ++ky) {
#pragma unroll
    for (int kx = 0; kx < 3; ++kx) {
      int hh = h + ky - 1, ww = wq + kx - 1;
      if (hh >= 0 && hh < 64 && ww >= 0 && ww < 64)
        s += in[((long)b * LL + hh * 64 + ww) * 80 + ch] * w[ch * 9 + ky * 3 + kx];
    }
  }
  out[i] = s;
}

// Instance-norm stats for the second half channels of an NHWC tensor.
__global__ __launch_bounds__(256) void k_inststats(const float* __restrict__ x,
                                                   int C, int half,
                                                   float* __restrict__ mu,
                                                   float* __restrict__ var) {
  int nb = C - half;
  int b = blockIdx.x / nb, ch = half + blockIdx.x % nb;
  const float* p = x + (long)b * LL * C + ch;
  float s = 0.f, q = 0.f;
  for (int l = threadIdx.x; l < LL; l += 256) {
    float v = p[(long)l * C];
    s += v;
    q += v * v;
  }
  __shared__ float rs[256], rq[256];
  rs[threadIdx.x] = s;
  rq[threadIdx.x] = q;
  __syncthreads();
  for (int st = 128; st > 0; st >>= 1) {
    if (threadIdx.x < st) {
      rs[threadIdx.x] += rs[threadIdx.x + st];
      rq[threadIdx.x] += rq[threadIdx.x + st];
    }
    __syncthreads();
  }
  if (threadIdx.x == 0) {
    float m = rs[0] * (1.f / LL);
    mu[blockIdx.x] = m;
    var[blockIdx.x] = rq[0] * (1.f / LL) - m * m;
  }
}

__global__ void k_ibnapply(float* __restrict__ x, int C, int half,
                           const float* __restrict__ g, const float* __restrict__ bt,
                           const float* __restrict__ rm, const float* __restrict__ rv,
                           const float* __restrict__ imu,
                           const float* __restrict__ ivar) {
  long i = (long)blockIdx.x * blockDim.x + threadIdx.x;  // B*L*C
  if (i >= (long)BB * LL * C) return;
  int ch = (int)(i % C);
  long t = i / C;
  int b = (int)(t >> 12);
  float v = x[i];
  if (ch < half) {
    v = (v - rm[ch]) * rsqrtf(rv[ch] + 1e-5f) * g[ch] + bt[ch];
  } else {
    int idx = b * (C - half) + (ch - half);
    v = (v - imu[idx]) * rsqrtf(ivar[idx] + 1e-5f);
  }
  x[i] = fmaxf(v, 0.f);
}

__global__ void k_final(const float* __restrict__ OUTB, const float* __restrict__ X2d,
                        const float* __restrict__ f, float* __restrict__ out) {
  long i = (long)blockIdx.x * blockDim.x + threadIdx.x;  // B*320*L (NCHW)
  if (i >= (long)BB * 320 * LL) return;
  int l = (int)(i & (LL - 1));
  long t = i >> 12;
  int ch = (int)(t % 320);
  int b = (int)(t / 320);
  long nidx = ((long)b * LL + l) * 320 + ch;
  float o = OUTB[nidx];
  out[i] = (o * f[i] + o) + X2d[nidx];
}

// ---------------------------------------------------------------------------
extern "C" void kernel_launch(void* const* d_in, const int* in_sizes, int n_in,
                              void* d_out, int out_size, void* d_ws, size_t ws_size,
                              hipStream_t stream) {
  (void)in_sizes; (void)n_in; (void)out_size; (void)ws_size;
  const float* f = (const float*)d_in[0];
  const float* x = (const float*)d_in[1];
  const float* norm_g = (const float*)d_in[2];
  const float* norm_b = (const float*)d_in[3];
  const float* proj_w = (const float*)d_in[4];
  const float* proj_b = (const float*)d_in[5];
  const float* skip_scale = (const float*)d_in[6];
  const float* m_in_w = (const float*)d_in[7];
  const float* m_conv_w = (const float*)d_in[8];
  const float* m_conv_b = (const float*)d_in[9];
  const float* m_xproj_w = (const float*)d_in[10];
  const float* m_dt_w = (const float*)d_in[11];
  const float* m_dt_b = (const float*)d_in[12];
  const float* m_Alog = (const float*)d_in[13];
  const float* m_D = (const float*)d_in[14];
  const float* m_out_w = (const float*)d_in[15];
  const float* se_fc1_w = (const float*)d_in[16];
  const float* se_fc2_w = (const float*)d_in[17];
  const float* conv1_w = (const float*)d_in[18];
  const float* conv1_b = (const float*)d_in[19];
  const float* conv2_w = (const float*)d_in[20];
  const float* conv2_b = (const float*)d_in[21];
  const float* fc1_w = (const float*)d_in[22];
  const float* fc1_b = (const float*)d_in[23];
  const float* fc1_bn_g = (const float*)d_in[24];
  const float* fc1_bn_b = (const float*)d_in[25];
  const float* fc1_bn_rm = (const float*)d_in[26];
  const float* fc1_bn_rv = (const float*)d_in[27];
  const float* dw_w = (const float*)d_in[28];
  const float* dw_b = (const float*)d_in[29];
  const float* dw_bn_g = (const float*)d_in[30];
  const float* dw_bn_b = (const float*)d_in[31];
  const float* dw_bn_rm = (const float*)d_in[32];
  const float* dw_bn_rv = (const float*)d_in[33];
  const float* fc2_w = (const float*)d_in[34];
  const float* fc2_b = (const float*)d_in[35];
  const float* fc2_bn_g = (const float*)d_in[36];
  const float* fc2_bn_b = (const float*)d_in[37];
  const float* fc2_bn_rm = (const float*)d_in[38];
  const float* fc2_bn_rv = (const float*)d_in[39];
  float* out = (float*)d_out;

  // workspace layout (float elements)
  float* ws = (float*)d_ws;
  size_t off = 0;
  float* CN = ws + off;  off += (size_t)BB * LL * CC;        // raw NHWC c; reused for ln2 out
  float* XN = ws + off;  off += (size_t)BB * LL * CC;        // layernorm1 out
  float* XZ = ws + off;  off += (size_t)MB * LL * CC;        // [xpart|zpart]; xpart cols become y'
  float* XC = ws + off;  off += (size_t)MB * LL * DI;        // conv1d+silu out
  float* DBLB = ws + off; off += (size_t)MB * LL * DBL_LD;   // [dt|B|C], padded to 48
  float* ZB = ws + off;  off += (size_t)BB * LL * 320;       // z NHWC; reused for X2B/X2C
  float* FN = ws + off;  off += (size_t)BB * LL * 320;       // f NHWC
  float* YM = ws + off;  off += (size_t)BB * LL * CC;        // out_proj + skip
  float* OUTB = ws + off; off += (size_t)BB * LL * 320;      // proj out NHWC
  float* X2A = ws + off; off += (size_t)BB * LL * 320;       // conv1 out; reused for X2D
  float* AVG = ws + off; off += (size_t)BB * CC;
  float* GATE = ws + off; off += (size_t)BB * CC;
  float* IMU = ws + off; off += 2048;
  float* IVAR = ws + off; off += 2048;
  float* X2B = ZB;                      // [B,L,80]
  float* X2C = ZB + (size_t)BB * LL * 80;
  float* X2D = X2A;

  const long SL = (long)LL * CC;  // 4096*512

  auto gemm = [&](const float* A, long aS1, long aS2, int bmod, int lda,
                  const float* W, int ldw, float* C, long cS1, long cS2, int ldc,
                  const float* bias, const float* skip, long sS1, long sS2,
                  int ldskip, const float* sc, int M, int N, int K, int acc,
                  int nb) {
    dim3 g((N + 63) / 64, M / 64, nb);
    gemm_wmma_f32<<<g, dim3(128), 0, stream>>>(A, aS1, aS2, bmod, lda, W, ldw, C,
                                               cS1, cS2, ldc, bias, skip, sS1,
                                               sS2, ldskip, sc, M, N, K, acc);
  };

  // 1) upsample x -> NHWC c (ungated)
  k_upsample<<<(int)(((long)BB * LL * CC + 255) / 256), 256, 0, stream>>>(x, CN);
  // 2) SE: channel means (of c == means of x), gate
  k_seavg<<<BB * CC, 256, 0, stream>>>(x, AVG);
  k_segate<<<BB, 512, 0, stream>>>(AVG, se_fc1_w, se_fc2_w, GATE);
  // 3) z = conv2(c) : GEMM on ungated c
  gemm(CN, SL, 0, BB, CC, conv2_w, CC, ZB, (long)LL * 320, 0, 320, conv2_b,
       nullptr, 0, 0, 0, nullptr, LL, 320, CC, 0, BB);
  // 4) layernorm1 on gated c
  k_layernorm<<<BB * LL, 256, 0, stream>>>(CN, GATE, norm_g, norm_b, XN);
  // 5) mamba in_proj (chunk views: mb = q*8 + b)
  gemm(XN, SL, 128, BB, CC, m_in_w, 128, XZ, SL, 8 * SL, CC, nullptr, nullptr, 0,
       0, 0, nullptr, LL, CC, 128, 0, MB);
  // 6) causal depthwise conv1d + silu
  k_mconv<<<(int)(((long)MB * LL * DI + 255) / 256), 256, 0, stream>>>(
      XZ, m_conv_w, m_conv_b, XC);
  // 7) x_proj -> [dt|B|C] (N=40, padded ld 48)
  gemm(XC, (long)LL * DI, 0, MB, DI, m_xproj_w, DI, DBLB, (long)LL * DBL_LD, 0,
       DBL_LD, nullptr, nullptr, 0, 0, 0, nullptr, LL, 40, DI, 0, MB);
  // 8) selective scan (fuses delta, D-skip, silu(z) gate); writes xpart cols
  k_scan<<<MB, DI, 0, stream>>>(XC, DBLB, XZ, m_dt_w, m_dt_b, m_Alog, m_D, XZ);
  // 9) out_proj + skip_scale*chunks, scattered back into [B,L,512]
  gemm(XZ, SL, 8 * SL, BB, CC, m_out_w, DI, YM, SL, 128, CC, nullptr, XN, SL,
       128, CC, skip_scale, LL, 128, DI, 0, MB);
  // 10) layernorm2 (reuse CN)
  k_layernorm<<<BB * LL, 256, 0, stream>>>(YM, nullptr, norm_g, norm_b, CN);
  // 11) proj -> OUT NHWC [B,L,320]
  gemm(CN, SL, 0, BB, CC, proj_w, CC, OUTB, (long)LL * 320, 0, 320, proj_b,
       nullptr, 0, 0, 0, nullptr, LL, 320, CC, 0, BB);
  // 12) f -> NHWC
  k_fnhwc<<<(int)(((long)BB * LL * 320 + 255) / 256), 256, 0, stream>>>(f, FN);
  // 13) conv1 on concat(z,f): two K=320 passes
  gemm(ZB, (long)LL * 320, 0, BB, 320, conv1_w, 640, X2A, (long)LL * 320, 0, 320,
       conv1_b, nullptr, 0, 0, 0, nullptr, LL, 320, 320, 0, BB);
  gemm(FN, (long)LL * 320, 0, BB, 320, conv1_w + 320, 640, X2A, (long)LL * 320, 0,
       320, nullptr, nullptr, 0, 0, 0, nullptr, LL, 320, 320, 1, BB);
  // 14) fc1 -> 80ch, ibnorm(40) + relu
  gemm(X2A, (long)LL * 320, 0, BB, 320, fc1_w, 320, X2B, (long)LL * 80, 0, 80,
       fc1_b, nullptr, 0, 0, 0, nullptr, LL, 80, 320, 0, BB);
  k_inststats<<<BB * 40, 256, 0, stream>>>(X2B, 80, 40, IMU, IVAR);
  k_ibnapply<<<(int)(((long)BB * LL * 80 + 255) / 256), 256, 0, stream>>>(
      X2B, 80, 40, fc1_bn_g, fc1_bn_b, fc1_bn_rm, fc1_bn_rv, IMU, IVAR);
  // 15) depthwise 3x3, ibnorm(40) + relu
  k_dwconv<<<(int)(((long)BB * LL * 80 + 255) / 256), 256, 0, stream>>>(
      X2B, dw_w, dw_b, X2C);
  k_inststats<<<BB * 40, 256, 0, stream>>>(X2C, 80, 40, IMU, IVAR);
  k_ibnapply<<<(int)(((long)BB * LL * 80 + 255) / 256), 256, 0, stream>>>(
      X2C, 80, 40, dw_bn_g, dw_bn_b, dw_bn_rm, dw_bn_rv, IMU, IVAR);
  // 16) fc2 -> 320ch, ibnorm(160) + relu
  gemm(X2C, (long)LL * 80, 0, BB, 80, fc2_w, 80, X2D, (long)LL * 320, 0, 320,
       fc2_b, nullptr, 0, 0, 0, nullptr, LL, 320, 80, 0, BB);
  k_inststats<<<BB * 160, 256, 0, stream>>>(X2D, 320, 160, IMU, IVAR);
  k_ibnapply<<<(int)(((long)BB * LL * 320 + 255) / 256), 256, 0, stream>>>(
      X2D, 320, 160, fc2_bn_g, fc2_bn_b, fc2_bn_rm, fc2_bn_rv, IMU, IVAR);
  // 17) final: X1 + X2, back to NCHW
  k_final<<<(int)(((long)BB * 320 * LL + 255) / 256), 256, 0, stream>>>(
      OUTB, X2D, f, out);
}